// GNNEdgeTypeClassifier_21852793602865
// MI455X (gfx1250) — compile-verified
//
#include <hip/hip_runtime.h>
#include <hip/hip_bf16.h>

typedef float v2f __attribute__((ext_vector_type(2)));
typedef float v8f __attribute__((ext_vector_type(8)));

#define TPB 256

// ---------------- pad weights (30x30 -> 32x32, 5x30 -> 32x32) and biases ----
__global__ void pad_params_kernel(const float* W1, const float* W2, const float* W3,
                                  const float* Wout, const float* b1, const float* b2,
                                  const float* b3, const float* bout,
                                  float* Wp, float* bp) {
    int idx = blockIdx.x * blockDim.x + threadIdx.x;
    if (idx < 4 * 1024) {
        int wsel = idx >> 10;          // which weight matrix
        int r    = (idx >> 5) & 31;    // output-feature row
        int c    = idx & 31;           // input-feature col
        const float* Ws = (wsel == 0) ? W1 : (wsel == 1) ? W2 : (wsel == 2) ? W3 : Wout;
        int rows = (wsel == 3) ? 5 : 30;
        Wp[idx] = (r < rows && c < 30) ? Ws[r * 30 + c] : 0.0f;
    } else if (idx < 4 * 1024 + 4 * 32) {
        int t = idx - 4096;
        int bsel = t >> 5;
        int c    = t & 31;
        const float* bs = (bsel == 0) ? b1 : (bsel == 1) ? b2 : (bsel == 2) ? b3 : bout;
        int len = (bsel == 3) ? 5 : 30;
        bp[t] = (c < len) ? bs[c] : 0.0f;
    }
}

// ---------------- h0 = emb[x], padded to stride 32 --------------------------
__global__ void embed_kernel(const int* x, const float* emb, float* h, int n) {
    int idx = blockIdx.x * blockDim.x + threadIdx.x;
    if (idx >= n * 32) return;
    int node = idx >> 5;
    int d    = idx & 31;
    h[idx] = (d < 30) ? emb[x[node] * 30 + d] : 0.0f;
}

// ---------------- degree (source degree + self loop) ------------------------
__global__ void deg_init_kernel(float* deg, int n) {
    int v = blockIdx.x * blockDim.x + threadIdx.x;
    if (v < n) deg[v] = 1.0f;  // self loop
}
__global__ void deg_count_kernel(const int* row, float* deg, int E) {
    int e = blockIdx.x * blockDim.x + threadIdx.x;
    if (e < E) atomicAdd(&deg[row[e]], 1.0f);
}
__global__ void deg_inv_kernel(float* deg, int n) {
    int v = blockIdx.x * blockDim.x + threadIdx.x;
    if (v < n) deg[v] = 1.0f / deg[v];
}

// ---------------- agg init with self-loop contribution ----------------------
__global__ void selfloop_kernel(const float* h, const float* invdeg, float* agg, int n) {
    int idx = blockIdx.x * blockDim.x + threadIdx.x;
    if (idx >= n * 32) return;
    agg[idx] = h[idx] * invdeg[idx >> 5];
}

// ---------------- edge scatter: agg[col] += h[row] * invdeg[row] ------------
// one wave-lane per feature dim; 128B coalesced gather per edge, f32 atomics.
__global__ void scatter_kernel(const int* row, const int* col, const float* h,
                               const float* invdeg, float* agg, int E) {
    int idx  = blockIdx.x * blockDim.x + threadIdx.x;
    int e    = idx >> 5;
    int lane = idx & 31;
    if (e >= E || lane >= 30) return;
    int r = row[e];
    int c = col[e];
    atomicAdd(&agg[c * 32 + lane], h[r * 32 + lane] * invdeg[r]);
}

// ---------------- dense: out = [relu](in @ Wp^T + bp) via V_WMMA_F32_16X16X4_F32
// in: [n,32] padded; Wp: [32,32] row-major (out-feature major); bp: [32]
// Each wave computes one 16-node tile: two 16x16 output tiles, K=32 in 8 chunks.
__global__ void dense_wmma_kernel(const float* in, const float* __restrict__ Wp,
                                  const float* __restrict__ bp, float* out,
                                  int out_stride, int out_cols, int do_relu, int n) {
    int gtid = blockIdx.x * blockDim.x + threadIdx.x;
    int wave = gtid >> 5;
    int lane = threadIdx.x & 31;
    int base = wave * 16;
    if (base >= n) return;              // wave-uniform: EXEC all-ones inside

    int m    = lane & 15;               // A row / B col / C col
    int half = lane >> 4;               // K sub-slot select

    v8f c0 = {};                        // output cols 0..15
    v8f c1 = {};                        // output cols 16..31

    const float* arow = in + (size_t)(base + m) * 32 + 2 * half;
#pragma unroll
    for (int kk = 0; kk < 8; ++kk) {
        // A frag: A[m][4kk + 2*half + {0,1}]
        v2f a = *(const v2f*)(arow + 4 * kk);
        // B frag tile j: B[k][n] = Wp[16j+n][k]; this lane: n=m, k=4kk+2*half+{0,1}
        v2f b0 = *(const v2f*)(Wp + (size_t)m * 32 + 4 * kk + 2 * half);
        v2f b1 = *(const v2f*)(Wp + (size_t)(16 + m) * 32 + 4 * kk + 2 * half);
        c0 = __builtin_amdgcn_wmma_f32_16x16x4_f32(false, a, false, b0,
                                                   (short)0, c0, false, false);
        c1 = __builtin_amdgcn_wmma_f32_16x16x4_f32(false, a, false, b1,
                                                   (short)0, c1, false, false);
    }

    float bias0 = bp[m];
    float bias1 = bp[16 + m];
#pragma unroll
    for (int v = 0; v < 8; ++v) {
        int r = v + 8 * half;           // node row within tile
        float x0 = c0[v] + bias0;
        float x1 = c1[v] + bias1;
        if (do_relu) { x0 = fmaxf(x0, 0.0f); x1 = fmaxf(x1, 0.0f); }
        size_t rowoff = (size_t)(base + r) * out_stride;
        if (m < out_cols)        out[rowoff + m]      = x0;
        if (16 + m < out_cols)   out[rowoff + 16 + m] = x1;
    }
}

// ---------------------------------------------------------------------------
extern "C" void kernel_launch(void* const* d_in, const int* in_sizes, int n_in,
                              void* d_out, int out_size, void* d_ws, size_t ws_size,
                              hipStream_t stream) {
    const int*   x    = (const int*)d_in[0];
    const int*   ei   = (const int*)d_in[1];    // [2, E]
    const float* emb  = (const float*)d_in[2];
    const float* W1   = (const float*)d_in[3];
    const float* b1   = (const float*)d_in[4];
    const float* W2   = (const float*)d_in[5];
    const float* b2   = (const float*)d_in[6];
    const float* W3   = (const float*)d_in[7];
    const float* b3   = (const float*)d_in[8];
    const float* Wout = (const float*)d_in[9];
    const float* bout = (const float*)d_in[10];
    float* out = (float*)d_out;

    const int N = in_sizes[0];
    const int E = in_sizes[1] / 2;
    const int* row = ei;       // edge_index[0]
    const int* col = ei + E;   // edge_index[1]

    // workspace layout (floats)
    float* hA     = (float*)d_ws;            // N*32
    float* hB     = hA + (size_t)N * 32;     // N*32
    float* invdeg = hB + (size_t)N * 32;     // N
    float* Wp     = invdeg + N;              // 4*1024
    float* bp     = Wp + 4 * 1024;           // 4*32

    auto blocks = [](long t) { return (int)((t + TPB - 1) / TPB); };

    pad_params_kernel<<<blocks(4 * 1024 + 4 * 32), TPB, 0, stream>>>(
        W1, W2, W3, Wout, b1, b2, b3, bout, Wp, bp);

    embed_kernel<<<blocks((long)N * 32), TPB, 0, stream>>>(x, emb, hA, N);

    deg_init_kernel<<<blocks(N), TPB, 0, stream>>>(invdeg, N);
    deg_count_kernel<<<blocks(E), TPB, 0, stream>>>(row, invdeg, E);
    deg_inv_kernel<<<blocks(N), TPB, 0, stream>>>(invdeg, N);

    int dense_threads_n = ((N + 15) / 16) * 32;

    float* cur = hA;
    float* nxt = hB;
    for (int layer = 0; layer < 3; ++layer) {
        selfloop_kernel<<<blocks((long)N * 32), TPB, 0, stream>>>(cur, invdeg, nxt, N);
        scatter_kernel<<<blocks((long)E * 32), TPB, 0, stream>>>(row, col, cur, invdeg, nxt, E);
        dense_wmma_kernel<<<blocks(dense_threads_n), TPB, 0, stream>>>(
            nxt, Wp + layer * 1024, bp + layer * 32, nxt, /*stride=*/32,
            /*cols=*/32, /*relu=*/1, N);
        float* t = cur; cur = nxt; nxt = t;
    }

    // head: [N,5] = cur @ Wout^T + bout (padded into Wp[3], bp[3])
    dense_wmma_kernel<<<blocks(dense_threads_n), TPB, 0, stream>>>(
        cur, Wp + 3 * 1024, bp + 3 * 32, out, /*stride=*/5, /*cols=*/5, /*relu=*/0, N);
}